// CNN_PHMM_VAE_78288663872511
// MI455X (gfx1250) — compile-verified
//
#include <hip/hip_runtime.h>
#include <stdint.h>

// ---------------------------------------------------------------------------
// pHMM-VAE loss for MI455X (gfx1250, wave32).
// Layout: 1 lane = 1 batch element. 64 threads (2 waves) per block, 64 blocks.
// Per-block staging of a/e/x tiles into LDS via the Tensor Data Mover (TDM),
// padded row strides to avoid LDS bank conflicts. State fM/fI/fD lives in
// VGPRs (fully unrolled k loops). Final deterministic reduction uses
// v_wmma_f32_16x16x4_f32 as a matrix row-sum.
// ---------------------------------------------------------------------------

typedef unsigned int u32x4 __attribute__((ext_vector_type(4)));
typedef int          i32x8 __attribute__((ext_vector_type(8)));
typedef int          i32x4 __attribute__((ext_vector_type(4)));
typedef float        f32x2 __attribute__((ext_vector_type(2)));
typedef float        f32x8 __attribute__((ext_vector_type(8)));

#define NEGV  (-100.0f)
#define LOGQ  (-1.3862943611198906f)   // log(0.25)
// transition indices: M2M=0, M2I=1, M2D=2, I2M=3, I2I=4, D2M=5, D2D=6

__device__ __forceinline__ uint32_t lds_addr_of(const void* p) {
  // Flat LDS-aperture addresses map to LDS via addr[31:0] (ISA §10.2).
  return (uint32_t)(uintptr_t)p;
}

__device__ __forceinline__ float lse2(float x, float y) {
  float m = fmaxf(x, y);
  float n = fminf(x, y);
  return m + __logf(1.0f + __expf(n - m));
}

__device__ __forceinline__ float lse3(float x, float y, float z) {
  float m = fmaxf(fmaxf(x, y), z);
  return m + __logf(__expf(x - m) + __expf(y - m) + __expf(z - m));
}

// 2D TDM load: `rows` rows of `dim0` 4-byte elements, contiguous in global
// (row stride == dim0).  Optional LDS padding: after every 2^(pad_iv+1)
// dwords stored, skip (pad_amt+1) dwords in LDS (-> conflict-free strides).
__device__ __forceinline__ void tdm_load_2d(uint32_t lds_addr, const void* gptr,
                                            uint32_t dim0, uint32_t rows,
                                            bool pad_en, uint32_t pad_iv,
                                            uint32_t pad_amt) {
  uint64_t ga = (uint64_t)(uintptr_t)gptr;
  u32x4 g0;
  g0[0] = 1u;                                           // count = 1 (valid D#)
  g0[1] = lds_addr;                                     // lds_addr (bytes)
  g0[2] = (uint32_t)ga;                                 // global_addr[31:0]
  g0[3] = (uint32_t)((ga >> 32) & 0x1FFFFFFu) | (2u << 30); // addr[56:32] | type=2

  uint32_t dw0 = (2u << 16);                            // data_size = 4 bytes
  if (pad_en) dw0 |= (1u << 20) | (pad_iv << 22) | (pad_amt << 25);
  i32x8 g1;
  g1[0] = (int)dw0;                                     // mask=0, size, pad ctl
  g1[1] = (int)((dim0 & 0xFFFFu) << 16);                // abar=0 | tensor_dim0.lo
  g1[2] = (int)(((dim0 >> 16) & 0xFFFFu) | ((rows & 0xFFFFu) << 16)); // d0.hi|d1.lo
  g1[3] = (int)(((rows >> 16) & 0xFFFFu) | ((dim0 & 0xFFFFu) << 16)); // d1.hi|tile0
  g1[4] = (int)(rows & 0xFFFFu);                        // tile_dim1 | tile_dim2=0
  g1[5] = (int)dim0;                                    // tensor_dim0_stride.lo
  g1[6] = 0;                                            // stride0.hi | stride1.lo
  g1[7] = 0;                                            // stride1.hi
  i32x4 z4 = {0, 0, 0, 0};                              // groups 2/3 unused (2D)
  i32x8 z8 = {0, 0, 0, 0, 0, 0, 0, 0};
  // amdgpu-toolchain (clang-23) 6-arg form: (g0, g1, g2, g3, extra, cpol)
  __builtin_amdgcn_tensor_load_to_lds(g0, g1, z4, z4, z8, 0);
}

__global__ __launch_bounds__(64)
void phmm_vae_kernel(const int* __restrict__ xg, const float* __restrict__ ag,
                     const float* __restrict__ eg, const float* __restrict__ mug,
                     const float* __restrict__ lvg, float* __restrict__ partial) {
  extern __shared__ float smem[];
  float* shA = smem;                    // 64 rows x 455 f32 (455 % 64 = 7: conflict-free)
  float* shE = shA + 64 * 455;          // 64 rows x 257 f32 (TDM-padded: 256+1)
  int*   shX = (int*)(shE + 64 * 257);  // 64 rows x 129 i32 (TDM-padded: 128+1)

  const int t  = threadIdx.x;
  const int b0 = blockIdx.x * 64;

  if (t == 0) {
    tdm_load_2d(lds_addr_of(shA), ag + (size_t)b0 * 455, 455, 64, false, 0, 0);
    tdm_load_2d(lds_addr_of(shE), eg + (size_t)b0 * 256, 256, 64, true, 7, 0); // +1/256dw
    tdm_load_2d(lds_addr_of(shX), xg + (size_t)b0 * 128, 128, 64, true, 6, 0); // +1/128dw
  }
  __builtin_amdgcn_s_wait_tensorcnt(0);
  __syncthreads();

  const float* A = shA + t * 455;   // A[k*7 + c]
  const float* E = shE + t * 257;   // E[k*4 + sym]
  const int*   X = shX + t * 129;

  float fM[65], fI[65], fD[65], fMn[65];
#pragma unroll
  for (int k = 0; k < 65; ++k) { fM[k] = NEGV; fI[k] = NEGV; }
  fM[0] = 0.0f;
  fD[0] = NEGV;
#pragma unroll
  for (int k = 1; k <= 64; ++k)
    fD[k] = lse2(A[(k - 1) * 7 + 2] + fM[k - 1], A[(k - 1) * 7 + 6] + fD[k - 1]);

#pragma unroll 1
  for (int l = 0; l < 128; ++l) {
    const int xl = X[l];
    fMn[0] = NEGV;
#pragma unroll
    for (int k = 0; k < 64; ++k) {
      float p = lse3(A[k * 7 + 0] + fM[k], A[k * 7 + 3] + fI[k],
                     A[k * 7 + 5] + fD[k]);
      fMn[k + 1] = E[k * 4 + xl] + p;
    }
#pragma unroll
    for (int k = 0; k < 65; ++k)
      fI[k] = LOGQ + lse2(A[k * 7 + 1] + fM[k], A[k * 7 + 4] + fI[k]);
    fD[0] = NEGV;
#pragma unroll
    for (int k = 1; k <= 64; ++k)
      fD[k] = lse2(A[(k - 1) * 7 + 2] + fMn[k - 1], A[(k - 1) * 7 + 6] + fD[k - 1]);
#pragma unroll
    for (int k = 0; k < 65; ++k) fM[k] = fMn[k];
  }

  const float fin = lse3(fM[64] + A[64 * 7 + 0], fI[64] + A[64 * 7 + 3],
                         fD[64] + A[64 * 7 + 5]);
  float v = -fin;                                        // -log P(x)

  const float* mu = mug + (size_t)(b0 + t) * 16;
  const float* lg = lvg + (size_t)(b0 + t) * 16;
  float kld = 0.0f;
#pragma unroll
  for (int j = 0; j < 16; ++j) {
    float m = mu[j], g = lg[j];
    kld += 1.0f + g - m * m - __expf(g);
  }
  v += -0.5f * kld;

  // block reduction: 2 waves of 32
  __shared__ float red[2];
#pragma unroll
  for (int off = 16; off > 0; off >>= 1) v += __shfl_down(v, off);
  if ((t & 31) == 0) red[t >> 5] = v;
  __syncthreads();
  if (t == 0) partial[blockIdx.x] = red[0] + red[1];
}

// Deterministic final reduction of 64 block partials using
// v_wmma_f32_16x16x4_f32 with B = ones: D[m,n] = sum_k A[m,k].
__global__ __launch_bounds__(32)
void phmm_reduce_kernel(const float* __restrict__ partial, float* __restrict__ out) {
  const int lane = threadIdx.x;
  // A (16x4 f32): lane<16 holds A[lane,0..1]; lane>=16 holds A[lane-16,2..3].
  const int i0 = (lane & 15) * 4 + ((lane >> 4) << 1);
  f32x2 Aop;  Aop[0] = partial[i0];  Aop[1] = partial[i0 + 1];
  f32x2 Bop;  Bop[0] = 1.0f;         Bop[1] = 1.0f;
  f32x8 C = {0.f, 0.f, 0.f, 0.f, 0.f, 0.f, 0.f, 0.f};
  f32x8 D = __builtin_amdgcn_wmma_f32_16x16x4_f32(false, Aop, false, Bop,
                                                  (short)0, C, false, false);
  // lane 0 holds D[0..7, 0] (rowsums 0..7); lane 16 holds D[8..15, 0].
  float s = D[0] + D[1] + D[2] + D[3] + D[4] + D[5] + D[6] + D[7];
  s += __shfl_down(s, 16);
  if (lane == 0) out[0] = s * (1.0f / 4096.0f);
}

extern "C" void kernel_launch(void* const* d_in, const int* in_sizes, int n_in,
                              void* d_out, int out_size, void* d_ws, size_t ws_size,
                              hipStream_t stream) {
  (void)in_sizes; (void)n_in; (void)out_size; (void)ws_size;
  const int*   x   = (const int*)d_in[0];    // (4096,128) int32
  const float* a   = (const float*)d_in[1];  // (4096,65,7) f32
  const float* e   = (const float*)d_in[2];  // (4096,64,4) f32
  const float* mus = (const float*)d_in[3];  // (4096,16)  f32
  const float* lv  = (const float*)d_in[4];  // (4096,16)  f32
  float* partial = (float*)d_ws;             // 64 block partials
  float* out = (float*)d_out;

  const size_t shbytes = (size_t)(64 * 455 + 64 * 257 + 64 * 129) * 4; // ~210 KB LDS
  phmm_vae_kernel<<<64, 64, shbytes, stream>>>(x, a, e, mus, lv, partial);
  phmm_reduce_kernel<<<1, 32, 0, stream>>>(partial, out);
}